// RealGCN_23072564314519
// MI455X (gfx1250) — compile-verified
//
#include <hip/hip_runtime.h>
#include <hip/hip_bf16.h>

typedef float v2f __attribute__((ext_vector_type(2)));
typedef float v8f __attribute__((ext_vector_type(8)));

#define N_NODES 50000

// ---------------------------------------------------------------------------
// Zero-fill
// ---------------------------------------------------------------------------
__global__ void gcn_zero_kernel(float* __restrict__ p, int n) {
    int i = blockIdx.x * blockDim.x + threadIdx.x;
    if (i < n) p[i] = 0.0f;
}

// ---------------------------------------------------------------------------
// Degree count: deg[dst[e]] += 1  (deg buffer pre-zeroed)
// ---------------------------------------------------------------------------
__global__ void gcn_degree_kernel(const int* __restrict__ dst, float* __restrict__ deg, int E) {
    int e = blockIdx.x * blockDim.x + threadIdx.x;
    if (e < E) atomicAdd(&deg[dst[e]], 1.0f);
}

// dinv[i] = (deg[i] + 1)^-0.5   (in place on the deg buffer)
__global__ void gcn_dinv_kernel(float* __restrict__ deg, int n) {
    int i = blockIdx.x * blockDim.x + threadIdx.x;
    if (i < n) deg[i] = rsqrtf(deg[i] + 1.0f);
}

// ---------------------------------------------------------------------------
// Dense GEMM C[M,N] = A[M,K] * B[K,N] via V_WMMA_F32_16X16X4_F32.
// One wave (32 lanes) computes one 16x16 tile; blockDim = (32, N/16).
// M=50000 (=3125*16), K=128, N in {128,64} -> no boundary handling needed.
// ---------------------------------------------------------------------------
__global__ __launch_bounds__(256) void gcn_gemm_wmma_kernel(
    const float* __restrict__ A, const float* __restrict__ B,
    float* __restrict__ C, int M, int K, int N)
{
    const int lane  = threadIdx.x;      // 0..31, one wave per y-row
    const int l15   = lane & 15;
    const int half  = lane >> 4;        // 0: lanes 0-15, 1: lanes 16-31
    const int mbase = blockIdx.x * 16;
    const int nbase = threadIdx.y * 16;

    const float* arow = A + (size_t)(mbase + l15) * K;   // A row for this lane
    v8f acc = {};

    for (int k = 0; k < K; k += 4) {
        const int ka = k + 2 * half;
        v2f a, b;
        // A 16x4 f32 layout: lane m=l15; VGPR0 holds K=2*half, VGPR1 K=2*half+1
        a.x = arow[ka + 0];
        a.y = arow[ka + 1];
        // B 4x16 f32 layout (mirror): col n=l15; rows 2*half and 2*half+1
        b.x = B[(size_t)(ka + 0) * N + nbase + l15];
        b.y = B[(size_t)(ka + 1) * N + nbase + l15];
        acc = __builtin_amdgcn_wmma_f32_16x16x4_f32(
            /*neg_a=*/false, a, /*neg_b=*/false, b,
            /*c_mod=*/(short)0, acc, /*reuse_a=*/false, /*reuse_b=*/false);
    }

    // C/D 16x16 f32 layout: element r -> row mbase + r + 8*half, col nbase + l15
    #pragma unroll
    for (int r = 0; r < 8; ++r) {
        C[(size_t)(mbase + r + 8 * half) * N + nbase + l15] = acc[r];
    }
}

// ---------------------------------------------------------------------------
// Edge aggregation: agg[dst] += dinv[src]*dinv[dst] * h[src]
// One thread handles 4 consecutive features of one edge (float4 gather).
// ---------------------------------------------------------------------------
__global__ void gcn_scatter_kernel(const float* __restrict__ h,
                                   const int* __restrict__ src,
                                   const int* __restrict__ dst,
                                   const float* __restrict__ dinv,
                                   float* __restrict__ agg, int E, int F)
{
    const int quads = F >> 2;
    int t = blockIdx.x * blockDim.x + threadIdx.x;
    if (t >= E * quads) return;
    const int e = t / quads;
    const int q = t - e * quads;
    const int s = src[e];
    const int d = dst[e];
    const float norm = dinv[s] * dinv[d];
    const float4 hv = *(const float4*)(h + (size_t)s * F + 4 * q);
    float* out = agg + (size_t)d * F + 4 * q;
    atomicAdd(out + 0, norm * hv.x);
    atomicAdd(out + 1, norm * hv.y);
    atomicAdd(out + 2, norm * hv.z);
    atomicAdd(out + 3, norm * hv.w);
}

// ---------------------------------------------------------------------------
// Finalize: out[i,f] = agg[i,f] + dinv[i]^2 * h[i,f] + bias[f]  (+ReLU)
// In place (out == agg) is fine.
// ---------------------------------------------------------------------------
__global__ void gcn_finalize_kernel(float* __restrict__ agg,
                                    const float* __restrict__ h,
                                    const float* __restrict__ dinv,
                                    const float* __restrict__ bias,
                                    float* __restrict__ out,
                                    int n_nodes, int F, int do_relu)
{
    int idx = blockIdx.x * blockDim.x + threadIdx.x;
    if (idx >= n_nodes * F) return;
    const int i = idx / F;
    const int f = idx - i * F;
    const float di = dinv[i];
    float v = agg[idx] + di * di * h[idx] + bias[f];
    if (do_relu) v = fmaxf(v, 0.0f);
    out[idx] = v;
}

// ---------------------------------------------------------------------------
// Launch
// ---------------------------------------------------------------------------
extern "C" void kernel_launch(void* const* d_in, const int* in_sizes, int n_in,
                              void* d_out, int out_size, void* d_ws, size_t ws_size,
                              hipStream_t stream) {
    const float* x  = (const float*)d_in[0];   // [N, 128]
    const int*   ei = (const int*)d_in[1];     // [2, E]
    const float* W1 = (const float*)d_in[2];   // [128, 128]
    const float* b1 = (const float*)d_in[3];   // [128]
    const float* W2 = (const float*)d_in[4];   // [128, 64]
    const float* b2 = (const float*)d_in[5];   // [64]
    float* out = (float*)d_out;                // [N, 64]

    const int NN = N_NODES;
    const int F1 = 128, F2 = 64, K = 128;
    const int E  = in_sizes[1] / 2;
    const int* src = ei;
    const int* dst = ei + E;

    // Workspace layout (floats): dinv | h1 | agg1/x2 | h2   (~64.2 MB total)
    float* ws   = (float*)d_ws;
    float* dinv = ws;                              // NN (50048-slot pad keeps 16B align)
    float* h1   = ws + 50048;                      // NN*F1
    float* x2   = h1 + (size_t)NN * F1;            // NN*F1 (agg1, then relu'd input)
    float* h2   = x2 + (size_t)NN * F1;            // NN*F2

    const int TB = 256;
    #define NB(n) (((n) + TB - 1) / TB)

    // --- normalization: deg -> dinv ---
    gcn_zero_kernel<<<NB(NN), TB, 0, stream>>>(dinv, NN);
    gcn_degree_kernel<<<NB(E), TB, 0, stream>>>(dst, dinv, E);
    gcn_dinv_kernel<<<NB(NN), TB, 0, stream>>>(dinv, NN);

    // --- layer 1: h1 = x @ W1 ---
    gcn_gemm_wmma_kernel<<<dim3(NN / 16), dim3(32, F1 / 16), 0, stream>>>(x, W1, h1, NN, K, F1);
    // agg1 = scatter(norm * h1[src] -> dst)
    gcn_zero_kernel<<<NB(NN * F1), TB, 0, stream>>>(x2, NN * F1);
    gcn_scatter_kernel<<<NB(E * (F1 / 4)), TB, 0, stream>>>(h1, src, dst, dinv, x2, E, F1);
    // x2 = relu(agg1 + dinv^2 * h1 + b1)
    gcn_finalize_kernel<<<NB(NN * F1), TB, 0, stream>>>(x2, h1, dinv, b1, x2, NN, F1, 1);

    // --- layer 2: h2 = x2 @ W2 ---
    gcn_gemm_wmma_kernel<<<dim3(NN / 16), dim3(32, F2 / 16), 0, stream>>>(x2, W2, h2, NN, K, F2);
    gcn_zero_kernel<<<NB(NN * F2), TB, 0, stream>>>(out, NN * F2);
    gcn_scatter_kernel<<<NB(E * (F2 / 4)), TB, 0, stream>>>(h2, src, dst, dinv, out, E, F2);
    gcn_finalize_kernel<<<NB(NN * F2), TB, 0, stream>>>(out, h2, dinv, b2, out, NN, F2, 0);

    #undef NB
}